// IDWT_22282290332121
// MI455X (gfx1250) — compile-verified
//
#include <hip/hip_runtime.h>

// ---------------------------------------------------------------------------
// db2 IDWT synthesis (fused 4-subband), exact fp32 via V_WMMA_F32_16X16X4_F32.
//
// Polyphase derivation (per axis, from the reference's pad/dilate/crop):
//   y[2j+p] = f[1-p]*x[j] + f[3-p]*x[min(j+1,511)]
// 2-D: y[pr,pc](jr,jc) = sum_{c,dr,dc} FH[c][2dr+1-pr]*FW[c][2dc+1-pc]
//                                      * x[jr+dr, jc+dc, c]   (clamped high edge)
// Factor per input pixel: P[px][n] = sum_c x[px][c]*W[c][n], n=(dr,dc,pr,pc).
// Each P entry is consumed exactly once -> WMMA does ALL the multiplies.
//
// Roofline: 268 MB traffic @ 23.3 TB/s ~= 11.5 us floor; WMMA projections cut
// ALU work 4x vs direct (16 MAC per *input* pixel instead of per output).
// ---------------------------------------------------------------------------

typedef __attribute__((ext_vector_type(2))) float v2f;
typedef __attribute__((ext_vector_type(8))) float v8f;

// lpf = [h3,h2,h1,h0], hpf = [-h0,h1,-h2,h3] with h* the db2 coefficients.
#define L0 (-0.12940952255126037f)
#define L1 ( 0.22414386804201339f)
#define L2 ( 0.83651630373780790f)
#define L3 ( 0.48296291314453416f)
#define H0 (-0.48296291314453416f)
#define H1 ( 0.83651630373780790f)
#define H2 (-0.22414386804201339f)
#define H3 (-0.12940952255126037f)

// Reference quirk preserved: channel order [LL, LH, HL, HH] uses
// (fh, fw) pairs = [(lpf,lpf), (lpf,lpf), (lpf,hpf), (hpf,hpf)].
__device__ __constant__ float c_FH[4][4] = {
    {L0, L1, L2, L3}, {L0, L1, L2, L3}, {L0, L1, L2, L3}, {H0, H1, H2, H3}};
__device__ __constant__ float c_FW[4][4] = {
    {L0, L1, L2, L3}, {L0, L1, L2, L3}, {H0, H1, H2, H3}, {H0, H1, H2, H3}};

#define TJ 32            // tile height in j-space (input pixels)
#define TC 16            // tile width in j-space
#define PRR (TJ + 1)     // 33 projection rows (halo)
#define PCC (TC + 1)     // 17 projection cols (halo)
#define NPIX (PRR * PCC) // 561 useful pixels
#define NCHUNK ((NPIX + 15) / 16)   // 36 WMMAs per workgroup
#define NPIXPAD (NCHUNK * 16)       // 576: padded so tail stores need no guard
#define PITCH 20         // floats per pixel in LDS: 80 B, 16 B aligned, conflict-free

__global__ __launch_bounds__(256) void idwt_db2_wmma(
    const float* __restrict__ x, float* __restrict__ out) {
  __shared__ __align__(16) float Plds[NPIXPAD * PITCH]; // 46,080 B

  const int tid  = threadIdx.x;
  const int lane = tid & 31;
  const int wave = tid >> 5;
  const int jc0  = blockIdx.x * TC;
  const int jr0  = blockIdx.y * TJ;
  const int b    = blockIdx.z;

  // ---- B operand: W[c][n], n = dr*8 + dc*4 + pr*2 + pc -------------------
  // Assumed f32 4x16 B layout (mirrors documented A 16x4 / C-D layouts):
  //   N = lane%16 ; K(channel) = vgpr + 2*(lane/16)
  //   => lanes 0-15 hold channels {0,1}, lanes 16-31 hold {2,3}.
  const int n  = lane & 15;
  const int hi = lane >> 4;      // lane half
  const int c0 = hi << 1;        // channel base for this half
  const int dr = (n >> 3) & 1, dc = (n >> 2) & 1;
  const int pr = (n >> 1) & 1, pc = n & 1;
  const int ihh = 2 * dr + 1 - pr;
  const int iww = 2 * dc + 1 - pc;
  v2f Bw;
  Bw.x = c_FH[c0][ihh] * c_FW[c0][iww];
  Bw.y = c_FH[c0 + 1][ihh] * c_FW[c0 + 1][iww];

  // ---- Stage 1: projections via WMMA, P -> LDS ---------------------------
  // A (16x4 f32, documented layout): M=pixel (lane%16), K=channel.
  // Lane loads float2 of its half's channel pair (NHWC, C=4 innermost).
  // Tail chunk (flat 561..575): prow<=33, pcol<=16; the gr/gc<=511 clamps keep
  // the global read in-bounds and the results land in LDS padding (unused),
  // so no guards -> no EXEC-mask churn around the DS stores.
  for (int chunk = wave; chunk < NCHUNK; chunk += 8) { // wave-uniform: EXEC all 1s at WMMA
    const int flat = chunk * 16 + n;
    const int prow = flat / PCC;
    const int pcol = flat - prow * PCC;
    int gr = jr0 + prow; gr = gr > 511 ? 511 : gr; // high-edge clamp (symmetric pad)
    int gc = jc0 + pcol; gc = gc > 511 ? 511 : gc;
    const float* px = x + ((((size_t)b << 9) + (size_t)gr) * 512 + (size_t)gc) * 4
                        + (hi << 1);
    v2f A; A.x = px[0]; A.y = px[1];
    v8f C = {0.f, 0.f, 0.f, 0.f, 0.f, 0.f, 0.f, 0.f};
    v8f D = __builtin_amdgcn_wmma_f32_16x16x4_f32(
        /*neg_a=*/false, A, /*neg_b=*/false, Bw,
        /*c_mod=*/(short)0, C, /*reuse_a=*/false, /*reuse_b=*/false);
    // D layout: lane holds N=lane%16 for M = v + 8*(lane/16), v=0..7.
    // One base address + 8 immediate-offset ds_store_b32 (offsets 0..560 B).
    float* p = &Plds[(chunk * 16 + (hi << 3)) * PITCH + n];
#pragma unroll
    for (int v = 0; v < 8; v++) {
      p[v * PITCH] = D[v];
    }
  }
  __syncthreads();

  // ---- Stage 2: 4-term gather per output phase, coalesced float4 stores --
  // Thread -> j-row r = tid/8, j-cols s0, s0+1. n-group for (dr,dc) is the
  // float4 at offset dr*8+dc*4 of pixel (r+dr, s+dc); its 4 components are
  // the phases (pr,pc) = (0,0),(0,1),(1,0),(1,1).
  const int r  = tid >> 3;
  const int s0 = (tid & 7) << 1;

  float4 accA, accB;
#pragma unroll
  for (int k = 0; k < 2; k++) {
    const int s = s0 + k;
    const float4 t00 = *(const float4*)&Plds[((r    ) * PCC + s    ) * PITCH + 0];
    const float4 t01 = *(const float4*)&Plds[((r    ) * PCC + s + 1) * PITCH + 4];
    const float4 t10 = *(const float4*)&Plds[((r + 1) * PCC + s    ) * PITCH + 8];
    const float4 t11 = *(const float4*)&Plds[((r + 1) * PCC + s + 1) * PITCH + 12];
    float4 a;
    a.x = t00.x + t01.x + t10.x + t11.x;  // (pr=0,pc=0)
    a.y = t00.y + t01.y + t10.y + t11.y;  // (pr=0,pc=1)
    a.z = t00.z + t01.z + t10.z + t11.z;  // (pr=1,pc=0)
    a.w = t00.w + t01.w + t10.w + t11.w;  // (pr=1,pc=1)
    if (k == 0) accA = a; else accB = a;
  }

  const size_t orow = (size_t)(jr0 + r) * 2;
  const size_t ocol = (size_t)(jc0 + s0) * 2;   // multiple of 4 -> 16 B aligned
  float* o = out + (((size_t)b * 1024 + orow) * 1024 + ocol);
  float4 ev; ev.x = accA.x; ev.y = accA.y; ev.z = accB.x; ev.w = accB.y;
  float4 od; od.x = accA.z; od.y = accA.w; od.z = accB.z; od.w = accB.w;
  *(float4*)(o)        = ev;   // output row 2jr
  *(float4*)(o + 1024) = od;   // output row 2jr+1
}

extern "C" void kernel_launch(void* const* d_in, const int* in_sizes, int n_in,
                              void* d_out, int out_size, void* d_ws, size_t ws_size,
                              hipStream_t stream) {
  (void)in_sizes; (void)n_in; (void)d_ws; (void)ws_size; (void)out_size;
  const float* x = (const float*)d_in[0];   // [32, 512, 512, 4] f32
  float* out = (float*)d_out;               // [32, 1024, 1024, 1] f32
  dim3 grid(512 / TC, 512 / TJ, 32);        // (32, 16, 32)
  idwt_db2_wmma<<<grid, 256, 0, stream>>>(x, out);
}